// AttnSeqTimeDecayModel_42855183679655
// MI455X (gfx1250) — compile-verified
//
#include <hip/hip_runtime.h>
#include <math.h>

// ---------------- problem constants (match reference) ----------------
#define NHIST   200000
#define TOPIC   128
#define HID     128
#define KSEL    50
#define NTILES  (NHIST / 16)          // 12500 row-tiles of 16
#define CHUNK   2000                  // rows per partial-topk block
#define P2_BLOCKS (NHIST / CHUNK)     // 100
#define NCAND   (P2_BLOCKS * KSEL)    // 5000

typedef float v2f __attribute__((ext_vector_type(2)));
typedef float v8f __attribute__((ext_vector_type(8)));

// =====================================================================
// Kernel 1: alpha[n] = dot(vs[n,:], v)  via V_WMMA_F32_16X16X4_F32.
// One wave <-> one 16-row tile. A(16x4) per-lane layout (ISA 7.12.2):
//   lanes 0..15 : M=lane,    VGPR0/1 = K = k0, k0+1
//   lanes 16..31: M=lane-16, VGPR0/1 = K = k0+2, k0+3
// B(4x16) has B[k][n] = v[k] (constant over n), same K striping.
// D column N=0 lives in lane 0 (M=0..7) and lane 16 (M=8..15).
// Memory-bound: every vs element loaded once, coalesced b64 loads.
// =====================================================================
__global__ void alpha_wmma_kernel(const float* __restrict__ vs,
                                  const float* __restrict__ v,
                                  float* __restrict__ alpha)
{
    __shared__ __align__(16) float sv[TOPIC];
    const int tid = threadIdx.x;
    if (tid < TOPIC) sv[tid] = v[tid];
    __syncthreads();

    const int lane   = tid & 31;
    const int wave   = tid >> 5;
    const int wpb    = blockDim.x >> 5;
    const int gwave  = blockIdx.x * wpb + wave;
    const int nwaves = gridDim.x * wpb;

    const int mloc = lane & 15;
    const int koff = (lane >> 4) << 1;        // 0 for lanes 0-15, 2 for 16-31

    for (int tile = gwave; tile < NTILES; tile += nwaves) {
        const int row = tile * 16 + mloc;
        const float* __restrict__ rp = vs + (size_t)row * TOPIC;

        // hint: stream the next tile this wave will touch
        __builtin_prefetch(rp + (size_t)nwaves * 16 * TOPIC, 0, 0);

        v8f c = {0.f, 0.f, 0.f, 0.f, 0.f, 0.f, 0.f, 0.f};
        #pragma unroll
        for (int k0 = 0; k0 < TOPIC; k0 += 4) {
            v2f a = *(const v2f*)(rp + k0 + koff);     // A 16x4 slice
            v2f b = *(const v2f*)(&sv[k0 + koff]);     // B 4x16 (v replicated)
            c = __builtin_amdgcn_wmma_f32_16x16x4_f32(
                    /*neg_a=*/false, a, /*neg_b=*/false, b,
                    /*c_mod=*/(short)0, c, /*reuse_a=*/false, /*reuse_b=*/false);
        }
        // extract column N=0
        if (lane == 0) {
            #pragma unroll
            for (int r = 0; r < 8; ++r) alpha[tile * 16 + r] = c[r];
        } else if (lane == 16) {
            #pragma unroll
            for (int r = 0; r < 8; ++r) alpha[tile * 16 + 8 + r] = c[r];
        }
    }
}

// =====================================================================
// Kernel 2: per-block top-50 of a 2000-element chunk (LDS argmax x50).
// Any global top-50 element is necessarily in its chunk's local top-50.
// =====================================================================
__global__ void topk_partial_kernel(const float* __restrict__ alpha,
                                    float* __restrict__ cand_v,
                                    int*   __restrict__ cand_i)
{
    __shared__ float sval[CHUNK];
    __shared__ float rv[256];
    __shared__ int   ri[256];
    const int tid  = threadIdx.x;
    const int base = blockIdx.x * CHUNK;

    for (int i = tid; i < CHUNK; i += blockDim.x) sval[i] = alpha[base + i];
    __syncthreads();

    for (int k = 0; k < KSEL; ++k) {
        float best = -INFINITY; int bi = -1;
        for (int i = tid; i < CHUNK; i += blockDim.x) {
            float x = sval[i];
            if (x > best) { best = x; bi = i; }
        }
        rv[tid] = best; ri[tid] = bi;
        __syncthreads();
        for (int s = blockDim.x >> 1; s > 0; s >>= 1) {
            if (tid < s && rv[tid + s] > rv[tid]) { rv[tid] = rv[tid + s]; ri[tid] = ri[tid + s]; }
            __syncthreads();
        }
        if (tid == 0) {
            cand_v[blockIdx.x * KSEL + k] = rv[0];
            cand_i[blockIdx.x * KSEL + k] = base + ri[0];
            sval[ri[0]] = -INFINITY;
        }
        __syncthreads();
    }
}

// =====================================================================
// Kernel 3 (single block): merge candidates -> global top-50, then
// decay, softmax, attention-weighted sum of gathered hs rows, score
// head, and one GRU step. All tiny (O(KB) of traffic).
// =====================================================================
__global__ void finalize_kernel(const float* __restrict__ v,
                                const float* __restrict__ s,
                                const float* __restrict__ t,
                                const float* __restrict__ hs,
                                const float* __restrict__ ts,
                                const float* __restrict__ W_ih,
                                const float* __restrict__ b_ih,
                                const float* __restrict__ W_hh,
                                const float* __restrict__ b_hh,
                                const float* __restrict__ W_score,
                                const float* __restrict__ b_score,
                                const float* __restrict__ cand_v_g,
                                const int*   __restrict__ cand_i_g,
                                float* __restrict__ out)
{
    __shared__ float cval[NCAND];
    __shared__ int   cidx[NCAND];
    __shared__ float rv[256];
    __shared__ int   ri[256];
    __shared__ float tv[KSEL];
    __shared__ int   tix[KSEL];
    __shared__ float w[KSEL];
    __shared__ float red[256];
    __shared__ float attn[HID];
    __shared__ float xbuf[TOPIC + 1];
    __shared__ float h0s[HID];
    __shared__ float gi[3 * HID];
    __shared__ float gh[3 * HID];

    const int tid = threadIdx.x;

    for (int i = tid; i < NCAND; i += blockDim.x) {
        cval[i] = cand_v_g[i];
        cidx[i] = cand_i_g[i];
    }
    __syncthreads();

    // ---- global top-50 over 5000 candidates ----
    for (int k = 0; k < KSEL; ++k) {
        float best = -INFINITY; int bi = -1;
        for (int i = tid; i < NCAND; i += blockDim.x) {
            float x = cval[i];
            if (x > best) { best = x; bi = i; }
        }
        rv[tid] = best; ri[tid] = bi;
        __syncthreads();
        for (int sz = blockDim.x >> 1; sz > 0; sz >>= 1) {
            if (tid < sz && rv[tid + sz] > rv[tid]) { rv[tid] = rv[tid + sz]; ri[tid] = ri[tid + sz]; }
            __syncthreads();
        }
        if (tid == 0) {
            tv[k]  = rv[0];
            tix[k] = cidx[ri[0]];
            cval[ri[0]] = -INFINITY;
        }
        __syncthreads();
    }

    // ---- time decay + softmax over K=50 (serial, trivial) ----
    if (tid == 0) {
        const float DEC = 1.0f - 1e-7f;
        const float tt  = t[0];
        float mx = -INFINITY;
        for (int k = 0; k < KSEL; ++k) {
            float dt = tt - ts[tix[k]];
            float a  = tv[k] * powf(DEC, dt);
            tv[k] = a;
            if (a > mx) mx = a;
        }
        float sum = 0.f;
        for (int k = 0; k < KSEL; ++k) { float e = expf(tv[k] - mx); w[k] = e; sum += e; }
        float inv = 1.f / sum;
        for (int k = 0; k < KSEL; ++k) w[k] *= inv;
    }
    __syncthreads();

    // ---- attn_h = w @ hs[idx] ; stage x=[v,s] and h0 = hs[N-1] ----
    for (int h = tid; h < HID; h += blockDim.x) {
        float acc = 0.f;
        for (int k = 0; k < KSEL; ++k)
            acc += w[k] * hs[(size_t)tix[k] * HID + h];
        attn[h] = acc;
        h0s[h]  = hs[(size_t)(NHIST - 1) * HID + h];
    }
    if (tid < TOPIC) xbuf[tid] = v[tid];
    if (tid == 0)    xbuf[TOPIC] = s[0];
    __syncthreads();

    // ---- score = W_score . [v, attn_h] + b_score ----
    float part = 0.f;
    for (int i = tid; i < TOPIC + HID; i += blockDim.x) {
        float pv = (i < TOPIC) ? v[i] : attn[i - TOPIC];
        part += W_score[i] * pv;
    }
    red[tid] = part;
    __syncthreads();
    for (int sz = blockDim.x >> 1; sz > 0; sz >>= 1) {
        if (tid < sz) red[tid] += red[tid + sz];
        __syncthreads();
    }
    if (tid == 0) out[0] = red[0] + b_score[0];

    // ---- GRU gates: gi = W_ih x + b_ih ; gh = W_hh h0 + b_hh ----
    for (int r = tid; r < 3 * HID; r += blockDim.x) {
        float acc = b_ih[r];
        const float* Wr = W_ih + (size_t)r * (TOPIC + 1);
        for (int c = 0; c < TOPIC + 1; ++c) acc += Wr[c] * xbuf[c];
        gi[r] = acc;
        float acc2 = b_hh[r];
        const float* Wh = W_hh + (size_t)r * HID;
        for (int c = 0; c < HID; ++c) acc2 += Wh[c] * h0s[c];
        gh[r] = acc2;
    }
    __syncthreads();

    // ---- h_new (torch gate order r, z, n) ----
    for (int h = tid; h < HID; h += blockDim.x) {
        float rg = 1.f / (1.f + expf(-(gi[h] + gh[h])));
        float zg = 1.f / (1.f + expf(-(gi[HID + h] + gh[HID + h])));
        float ng = tanhf(gi[2 * HID + h] + rg * gh[2 * HID + h]);
        out[1 + h] = (1.f - zg) * ng + zg * h0s[h];
    }
}

// =====================================================================
extern "C" void kernel_launch(void* const* d_in, const int* in_sizes, int n_in,
                              void* d_out, int out_size, void* d_ws, size_t ws_size,
                              hipStream_t stream) {
    const float* v       = (const float*)d_in[0];
    const float* s       = (const float*)d_in[1];
    const float* t       = (const float*)d_in[2];
    const float* vs      = (const float*)d_in[3];
    const float* hs      = (const float*)d_in[4];
    const float* ts      = (const float*)d_in[5];
    const float* W_ih    = (const float*)d_in[6];
    const float* b_ih    = (const float*)d_in[7];
    const float* W_hh    = (const float*)d_in[8];
    const float* b_hh    = (const float*)d_in[9];
    const float* W_score = (const float*)d_in[10];
    const float* b_score = (const float*)d_in[11];
    float* out = (float*)d_out;

    // workspace layout
    float* alpha  = (float*)d_ws;          // NHIST floats
    float* cand_v = alpha + NHIST;         // NCAND floats
    int*   cand_i = (int*)(cand_v + NCAND);// NCAND ints

    alpha_wmma_kernel<<<512, 256, 0, stream>>>(vs, v, alpha);
    topk_partial_kernel<<<P2_BLOCKS, 256, 0, stream>>>(alpha, cand_v, cand_i);
    finalize_kernel<<<1, 256, 0, stream>>>(v, s, t, hs, ts,
                                           W_ih, b_ih, W_hh, b_hh,
                                           W_score, b_score,
                                           cand_v, cand_i, out);
}